// PROLoss_62517543960971
// MI455X (gfx1250) — compile-verified
//
#include <hip/hip_runtime.h>
#include <math.h>

typedef __attribute__((ext_vector_type(2))) float v2f;
typedef __attribute__((ext_vector_type(8))) float v8f;

#define EPSV           1e-8f
#define INV_TEMP       20.0f                 // 1 / 0.05
#define LOG2E          1.4426950408889634f
#define LN2            0.6931471805599453f
#define BLOCK          256                   // 8 wave32s
#define ROWS_PER_BLOCK 128                   // 16 rows per wave

// ---------------- kernel A: t_min = min(true_similarity) -> ws[0]
__global__ __launch_bounds__(BLOCK)
void pro_min_kernel(const float* __restrict__ tru, float* __restrict__ ws, int n) {
    __shared__ float red[BLOCK];
    float m = INFINITY;
    for (int i = threadIdx.x; i < n; i += BLOCK) m = fminf(m, tru[i]);
    red[threadIdx.x] = m;
    __syncthreads();
    for (int s = BLOCK / 2; s > 0; s >>= 1) {
        if (threadIdx.x < s) red[threadIdx.x] = fminf(red[threadIdx.x], red[threadIdx.x + s]);
        __syncthreads();
    }
    if (threadIdx.x == 0) ws[0] = red[0];
}

// ---------------- kernel B: per (16-row tile, j-chunk): masked max + sum-exp2.
// Logits are in base-2 units (pre-scaled by log2e) so exp is a bare v_exp_f32.
// Masking is branchless: masked-out elements get -inf logits, exp2(-inf)==0.
// The j-reduction runs on the matrix pipe via V_WMMA_F32_16X16X4_F32 with an
// all-ones B (D = row sums); two independent accumulators break the D->C chain.
// Loops use a uniform (scalar) trip counter so flow control stays on SALU and
// EXEC remains all-ones for the WMMAs.
__global__ __launch_bounds__(BLOCK)
void pro_main_kernel(const float* __restrict__ pred, const float* __restrict__ tru,
                     float* __restrict__ ws, int n, int jchunks, int chunkLen) {
    extern __shared__ float4 tp4s[];           // 16B-aligned dynamic LDS
    float2* tp = (float2*)tp4s;                // (t[j], p[j]*log2e/T) interleaved

    const int tid   = threadIdx.x;
    const int cid   = blockIdx.y;
    const int jbase = cid * chunkLen;

    for (int i = tid; i < chunkLen; i += BLOCK) {
        int j = jbase + i;
        tp[i] = make_float2(tru[j], pred[j] * (INV_TEMP * LOG2E));
    }
    __syncthreads();

    const int lane = tid & 31;
    const int wv   = tid >> 5;
    const int r    = lane & 15;                // row within the 16-row tile
    const int half = lane >> 4;                // K-half per the 16x4 f32 A layout
    const int row  = blockIdx.x * ROWS_PER_BLOCK + wv * 16 + r;

    const float tk = tru[row];
    const float NEG_INF = -INFINITY;
    const float4* __restrict__ tp4 = (const float4*)tp;

    // ---- pass 1: chunk-local masked max of l2 = p2[j] * (tk - t[j])
    float m0 = NEG_INF, m1 = NEG_INF;
    const int iters1 = chunkLen >> 2;          // 4 j-columns per iteration
#pragma unroll 4
    for (int it = 0; it < iters1; ++it) {      // uniform trip count -> SALU loop
        float4 q = tp4[half + 2 * it];         // one ds_load_b128: (t0,p0,t1,p1)
        float w0 = tk - q.x, w1 = tk - q.z;
        float c0 = (w0 > EPSV) ? q.y * w0 : NEG_INF;   // v_cndmask, no branch
        float c1 = (w1 > EPSV) ? q.w * w1 : NEG_INF;
        m0 = fmaxf(m0, c0);
        m1 = fmaxf(m1, c1);
    }
    float m = fmaxf(m0, m1);
    m = fmaxf(m, __shfl_xor(m, 16, 32));       // merge the two K-halves per row

    // ---- pass 2: chunk-local sum of 2^(l2 - m), reduced along j via WMMA
    v8f C0 = {}, C1 = {};
    v2f B; B.x = 1.0f; B.y = 1.0f;             // ones B-matrix -> D = row sums
    const int iters2 = chunkLen >> 3;          // 8 j-columns per iteration
#pragma unroll
    for (int it = 0; it < iters2; ++it) {
        {
            float4 q = tp4[half + 4 * it];
            float w0 = tk - q.x, w1 = tk - q.z;
            float a0 = (w0 > EPSV) ? __builtin_fmaf(q.y, w0, -m) : NEG_INF;
            float a1 = (w1 > EPSV) ? __builtin_fmaf(q.w, w1, -m) : NEG_INF;
            v2f A; A.x = __builtin_amdgcn_exp2f(a0); A.y = __builtin_amdgcn_exp2f(a1);
            C0 = __builtin_amdgcn_wmma_f32_16x16x4_f32(false, A, false, B,
                                                       (short)0, C0, false, false);
        }
        {
            float4 q = tp4[half + 4 * it + 2];
            float w0 = tk - q.x, w1 = tk - q.z;
            float a0 = (w0 > EPSV) ? __builtin_fmaf(q.y, w0, -m) : NEG_INF;
            float a1 = (w1 > EPSV) ? __builtin_fmaf(q.w, w1, -m) : NEG_INF;
            v2f A; A.x = __builtin_amdgcn_exp2f(a0); A.y = __builtin_amdgcn_exp2f(a1);
            C1 = __builtin_amdgcn_wmma_f32_16x16x4_f32(false, A, false, B,
                                                       (short)0, C1, false, false);
        }
    }
    v8f C = C0 + C1;                           // v_pk_add_f32 merge

    // C/D layout: VGPR v, lanes 0-15 -> M=v; lanes 16-31 -> M=8+v. All N columns
    // are identical (B==1), so pick the sum for row r via one bpermute.
    float x = C[lane & 7];
    int srcLane = (r < 8) ? r : (r + 8);
    float s = __shfl(x, srcLane, 32);

    if (half == 0) {                           // lanes 0-15 own rows 0-15
        float* Mpart = ws + 16;
        float* Spart = Mpart + n * jchunks;    // 32-bit indexing (n*jc < 2^31)
        Mpart[row * jchunks + cid] = m;
        Spart[row * jchunks + cid] = s;
    }
}

// ---------------- kernel C: merge chunk softmax stats (base-2), finalize loss
__global__ __launch_bounds__(BLOCK)
void pro_final_kernel(const float* __restrict__ pred, const float* __restrict__ tru,
                      const float* __restrict__ ws, float* __restrict__ out,
                      int n, int jchunks) {
    __shared__ float sT[BLOCK], sC[BLOCK];
    const float tmin = ws[0];
    const float* Mpart = ws + 16;
    const float* Spart = Mpart + n * jchunks;

    float total = 0.0f, count = 0.0f;
    for (int row = threadIdx.x; row < n; row += BLOCK) {
        float tk  = tru[row];
        float pk2 = pred[row] * (INV_TEMP * LOG2E);   // base-2 scaled logit slope
        bool  valid    = (tk - tmin) > EPSV;
        float weight   = valid ? (tk - tmin) : 0.0f;  // max gap = tk - t_min
        float logitPos = pk2 * weight;                // base-2 units

        float m = logitPos;
        for (int c = 0; c < jchunks; ++c)
            m = fmaxf(m, Mpart[row * jchunks + c]);
        if (!valid) m = 0.0f;

        float sum = __builtin_amdgcn_exp2f(logitPos - m);
        for (int c = 0; c < jchunks; ++c) {
            float mc = Mpart[row * jchunks + c];
            float sc = Spart[row * jchunks + c];
            sum += sc * __builtin_amdgcn_exp2f(mc - m);  // sc==0 when mc==-inf
        }
        // natural-log prob = ln2 * (base-2 log prob)
        float logProb = LN2 * (logitPos - m - __builtin_amdgcn_logf(sum));
        if (valid) { total += logProb; count += 1.0f; }
    }
    sT[threadIdx.x] = total; sC[threadIdx.x] = count;
    __syncthreads();
    for (int s = BLOCK / 2; s > 0; s >>= 1) {
        if (threadIdx.x < s) {
            sT[threadIdx.x] += sT[threadIdx.x + s];
            sC[threadIdx.x] += sC[threadIdx.x + s];
        }
        __syncthreads();
    }
    if (threadIdx.x == 0)
        out[0] = (sC[0] > 0.0f) ? (-sT[0] / fmaxf(sC[0], 1.0f)) : 0.0f;
}

extern "C" void kernel_launch(void* const* d_in, const int* in_sizes, int n_in,
                              void* d_out, int out_size, void* d_ws, size_t ws_size,
                              hipStream_t stream) {
    const float* pred = (const float*)d_in[0];   // predict_similarity
    const float* tru  = (const float*)d_in[1];   // true_similarity
    float* ws  = (float*)d_ws;
    float* out = (float*)d_out;
    const int n = in_sizes[0];                   // 8192

    // j-split for occupancy: largest power of two <= 8 whose partial buffers
    // fit the workspace (deterministic: ws_size is fixed across calls).
    int jc = 8;
    while (jc > 1 && (size_t)(16 + 2ull * (size_t)n * (size_t)jc) * sizeof(float) > ws_size)
        jc >>= 1;
    const int chunkLen  = n / jc;                // 8192/8 = 1024: multiple of 8
    const int rowBlocks = n / ROWS_PER_BLOCK;

    pro_min_kernel<<<1, BLOCK, 0, stream>>>(tru, ws, n);

    dim3 grid(rowBlocks, jc);
    size_t shbytes = (size_t)chunkLen * sizeof(float2);
    pro_main_kernel<<<grid, BLOCK, shbytes, stream>>>(pred, tru, ws, n, jc, chunkLen);

    pro_final_kernel<<<1, BLOCK, 0, stream>>>(pred, tru, ws, out, n, jc);
}